// ClassConditionalBatchNorm2d_76192719831905
// MI455X (gfx1250) — compile-verified
//
#include <hip/hip_runtime.h>
#include <math.h>
#include <stdint.h>

// ---------------------------------------------------------------------------
// ClassConditionalBatchNorm2d (eval path), MI455X / gfx1250
//
// Roofline: 2 FLOP per 8 bytes streamed -> pure HBM-bandwidth problem
// (411 MB @ 23.3 TB/s ~= 17.6 us). Strategy:
//   Kernel 1: fold stats into per-(b,c) {scale, shift}  (16384 float2, 128 KB)
//   Kernel 2: stream x via global_load_async_to_lds_b128 (CDNA5 async path,
//             ASYNCcnt), 4-deep LDS pipeline, fused FMA, NT 128-bit stores.
// ---------------------------------------------------------------------------

typedef float v4f __attribute__((ext_vector_type(4)));   // native vector type

#define BN_EPS        1e-5f
#define BN_EFF        0.3f     // min(ALPHA=0.3, 0.5)
#define BN_VAR_FLOOR  0.1f
#define BN_CNT_THRESH 100

constexpr int kB  = 64;
constexpr int kC  = 256;
constexpr int kHW = 56 * 56;            // 3136 floats per plane (12544 B)
constexpr int kWavesPerBlock = 8;       // 256 threads, wave32
constexpr int kBlocks        = 1024;    // 8192 waves total
constexpr int kChunksPerWave = 49;      // 49 * 128 floats = 6272 = 2 planes
// 8192 waves * 49 chunks * 128 floats = 51,380,224 = 64*256*56*56 (exact)

// ---------------------------------------------------------------------------
// Kernel 1: per-(b,c) scale/shift.  scale = w/sqrt(var+eps), shift = b - m*scale
// ---------------------------------------------------------------------------
__global__ __launch_bounds__(kC) void bn_stats_kernel(
    const int* __restrict__ labels,
    const float* __restrict__ gmean, const float* __restrict__ gvar,
    const float* __restrict__ cmean, const float* __restrict__ cvar,
    const int* __restrict__ counts,
    const float* __restrict__ weight, const float* __restrict__ bias,
    float2* __restrict__ ss)
{
  const int b = blockIdx.x;      // 0..63
  const int c = threadIdx.x;     // 0..255
  const int lab = labels[b];
  const bool use_class = (counts[lab] >= BN_CNT_THRESH);   // ALPHA != 0 at compile time

  const float gm = gmean[c];
  const float gv = gvar[c];
  float m, v;
  if (use_class) {
    const size_t o = (size_t)lab * kC + c;
    m = (1.0f - BN_EFF) * gm + BN_EFF * cmean[o];
    v = fmaxf((1.0f - BN_EFF) * gv + BN_EFF * cvar[o], BN_VAR_FLOOR);
  } else {
    m = gm;
    v = gv;
  }
  const float inv = 1.0f / sqrtf(v + BN_EPS);
  const float sc  = weight[c] * inv;
  const float sh  = fmaf(-m, sc, bias[c]);
  ss[b * kC + c] = make_float2(sc, sh);
}

// ---------------------------------------------------------------------------
// Kernel 2: streaming normalize. Each wave owns exactly 2 planes (6272 floats).
// x is staged HBM -> LDS with async b128 loads (ASYNCcnt), 4 buffers deep.
// ---------------------------------------------------------------------------

// LDS byte address of a shared-memory element, as the async-LDS ops want it.
// AS(3)->generic cast puts the LDS offset in the low 32 bits; the ptrtoint
// also makes the shared array escape so its loads can't be folded to undef.
__device__ __forceinline__ unsigned lds_addr(const void* p) {
  return (unsigned)(uintptr_t)p;
}

__global__ __launch_bounds__(256) void bn_apply_kernel(
    const float* __restrict__ x,
    const v4f* __restrict__ sspack,      // {sc0,sh0,sc1,sh1} per wave at index w
    float* __restrict__ out)
{
  // 8 waves * 4 buffers * 32 lanes * 16 B = 16 KB LDS per block
  __shared__ v4f smem[kWavesPerBlock * 4 * 32];

  const int lane      = threadIdx.x & 31;
  const int waveInBlk = threadIdx.x >> 5;
  const int w         = blockIdx.x * kWavesPerBlock + waveInBlk;   // 0..8191

  // Two planes' fused coefficients in one 16-B load (ss[2w], ss[2w+1]).
  const v4f co = sspack[w];   // co.x/co.y = plane 2w, co.z/co.w = plane 2w+1

  const size_t waveFloatBase = (size_t)w * (2 * kHW);              // 6272 * w
  const float* __restrict__ gsrc = x + waveFloatBase + lane * 4;
  float* __restrict__ gdst       = out + waveFloatBase + lane * 4;

  const int smemLane = waveInBlk * 128 + lane;   // this lane's slot, buffer 0

  // ---- prologue: issue chunks 0..2 into buffers 0..2 ----
  #pragma unroll
  for (int j = 0; j < 3; ++j) {
    const unsigned lds = lds_addr(&smem[smemLane + j * 32]);
    const float*   g   = gsrc + (size_t)j * 128;
    asm volatile("global_load_async_to_lds_b128 %0, %1, off"
                 :: "v"(lds), "v"(g) : "memory");
  }

  for (int i = 0; i < kChunksPerWave; ++i) {
    const int cur = i & 3;
    if (i + 3 < kChunksPerWave) {
      // Buffer (i+3)&3 was last DS-read at iteration i-1: async LDS writes are
      // unordered vs DS reads, so retire outstanding DS loads before reuse.
      asm volatile("s_wait_dscnt 0" ::: "memory");
      const unsigned lds = lds_addr(&smem[smemLane + ((i + 3) & 3) * 32]);
      const float*   g   = gsrc + (size_t)(i + 3) * 128;
      asm volatile("global_load_async_to_lds_b128 %0, %1, off"
                   :: "v"(lds), "v"(g) : "memory");
      // Async loads complete in order: <=3 outstanding => chunk i is in LDS.
      asm volatile("s_wait_asynccnt 3" ::: "memory");
    } else {
      asm volatile("s_wait_asynccnt 0" ::: "memory");
    }

    const v4f v = smem[smemLane + cur * 32];         // ds_load_b128

    // Plane select inside the wave's 2-plane span: local float index vs 3136.
    const int   local = i * 128 + lane * 4;          // 0..6268, float4-aligned
    const float sc = (local >= kHW) ? co.z : co.x;
    const float sh = (local >= kHW) ? co.w : co.y;

    v4f r;
    r.x = fmaf(v.x, sc, sh);
    r.y = fmaf(v.y, sc, sh);
    r.z = fmaf(v.z, sc, sh);
    r.w = fmaf(v.w, sc, sh);

    // 205 MB output stream: keep it out of L2's way with NT 128-bit stores.
    __builtin_nontemporal_store(r, (v4f*)(gdst + (size_t)i * 128));
  }
}

// ---------------------------------------------------------------------------
extern "C" void kernel_launch(void* const* d_in, const int* in_sizes, int n_in,
                              void* d_out, int out_size, void* d_ws, size_t ws_size,
                              hipStream_t stream) {
  const float* x      = (const float*)d_in[0];
  const int*   labels = (const int*)  d_in[1];
  const float* gmean  = (const float*)d_in[2];
  const float* gvar   = (const float*)d_in[3];
  const float* cmean  = (const float*)d_in[4];
  const float* cvar   = (const float*)d_in[5];
  const int*   counts = (const int*)  d_in[6];
  const float* weight = (const float*)d_in[7];
  const float* bias   = (const float*)d_in[8];
  float* out = (float*)d_out;

  float2* ss = (float2*)d_ws;   // 16384 float2 = 128 KB scratch, rewritten fully

  bn_stats_kernel<<<kB, kC, 0, stream>>>(labels, gmean, gvar, cmean, cvar,
                                         counts, weight, bias, ss);
  bn_apply_kernel<<<kBlocks, kWavesPerBlock * 32, 0, stream>>>(
      x, (const v4f*)ss, out);
}